// LinearGaussianContinuum_37684043055756
// MI455X (gfx1250) — compile-verified
//
#include <hip/hip_runtime.h>
#include <hip/hip_bf16.h>
#include <math.h>

typedef __attribute__((ext_vector_type(2))) float v2f;
typedef __attribute__((ext_vector_type(8))) float v8f;

#define DIMN 128
#define STR  132            // LDS row stride (pad: 132 mod 64 banks -> stride-4 banks, conflict-free)
#define LOG2PI_F 1.8378770664093453f

// ---------------------------------------------------------------------------
// Precompute (once per launch, 1 workgroup, 128 threads):
//   L0 = chol(C0); M = L0^-1 C1 L0^-T; Jacobi: M = Q Lam Q^T; V = Q^T L0^-1
//   e = V (m1 - m0); slog = sum log diag(L0)
// A, Q, L0 all live in LDS (3 * 128*132 * 4B = 198 KB < 320 KB/WGP).
// ---------------------------------------------------------------------------
__global__ void lgc_precompute(const float* __restrict__ cov0,
                               const float* __restrict__ cov1,
                               const float* __restrict__ mean0,
                               const float* __restrict__ mean1,
                               float* __restrict__ Vg,     // 128*128
                               float* __restrict__ lamg,   // 128
                               float* __restrict__ eg,     // 128
                               float* __restrict__ slogg)  // 1
{
    extern __shared__ float sm[];
    float* A  = sm;                    // 128*STR : X, then M, then Jacobi work
    float* Q  = sm + DIMN * STR;       // 128*STR
    float* L0 = sm + 2 * DIMN * STR;   // 128*STR
    __shared__ int   pp[64], qq[64];
    __shared__ float cs[64], sn[64];

    const int tid = threadIdx.x;       // 128 threads

    for (int i = tid; i < DIMN * STR; i += 128) { L0[i] = 0.f; Q[i] = 0.f; }
    __syncthreads();

    // ---- Cholesky of cov0 -> L0 (left-looking) ----
    for (int k = 0; k < DIMN; ++k) {
        if (tid == 0) {
            float s = cov0[k * DIMN + k];
            for (int j = 0; j < k; ++j) { float l = L0[k * STR + j]; s -= l * l; }
            L0[k * STR + k] = sqrtf(fmaxf(s, 1e-20f));
        }
        __syncthreads();
        float dk = L0[k * STR + k];
        for (int i = k + 1 + tid; i < DIMN; i += 128) {
            float s = cov0[i * DIMN + k];
            for (int j = 0; j < k; ++j) s -= L0[i * STR + j] * L0[k * STR + j];
            L0[i * STR + k] = s / dk;
        }
        __syncthreads();
    }

    // ---- X = L0^-1 * C1  (thread owns column c; no cross-thread deps) ----
    {
        const int c = tid;
        for (int i = 0; i < DIMN; ++i) {
            float s = cov1[i * DIMN + c];
            for (int j = 0; j < i; ++j) s -= L0[i * STR + j] * A[j * STR + c];
            A[i * STR + c] = s / L0[i * STR + i];
        }
    }
    __syncthreads();
    // ---- M = X * L0^-T  (in place, thread owns row r) ----
    {
        const int r = tid;
        for (int i = 0; i < DIMN; ++i) {
            float s = A[r * STR + i];
            for (int j = 0; j < i; ++j) s -= A[r * STR + j] * L0[i * STR + j];
            A[r * STR + i] = s / L0[i * STR + i];
        }
    }
    __syncthreads();
    // symmetrize
    for (int idx = tid; idx < DIMN * DIMN; idx += 128) {
        int i = idx >> 7, j = idx & 127;
        if (i < j) {
            float m = 0.5f * (A[i * STR + j] + A[j * STR + i]);
            A[i * STR + j] = m; A[j * STR + i] = m;
        }
    }
    if (tid < DIMN) Q[tid * STR + tid] = 1.f;
    __syncthreads();

    // ---- Parallel cyclic Jacobi: A <- Q^T M Q (fixed 10 sweeps, deterministic) ----
    for (int sweep = 0; sweep < 10; ++sweep) {
        for (int r = 0; r < 127; ++r) {
            if (tid < 64) {           // round-robin tournament pairing (fix 127)
                int p, q;
                if (tid == 0) { p = 127; q = r; }
                else { p = (r + tid) % 127; q = (r - tid + 127) % 127; }
                float app = A[p * STR + p], aqq = A[q * STR + q], apq = A[p * STR + q];
                float c = 1.f, s = 0.f;
                if (fabsf(apq) > 1e-30f) {
                    float th = (aqq - app) / (2.f * apq);
                    float tt = (th >= 0.f ? 1.f : -1.f) / (fabsf(th) + sqrtf(th * th + 1.f));
                    c = 1.f / sqrtf(1.f + tt * tt);
                    s = tt * c;
                }
                pp[tid] = p; qq[tid] = q; cs[tid] = c; sn[tid] = s;
            }
            __syncthreads();
            // rows: A <- J^T A   (pairs have disjoint rows)
            for (int idx = tid; idx < 64 * DIMN; idx += 128) {
                int k = idx >> 7, j = idx & 127;
                int p = pp[k], q = qq[k]; float c = cs[k], s = sn[k];
                float ap = A[p * STR + j], aq = A[q * STR + j];
                A[p * STR + j] = c * ap - s * aq;
                A[q * STR + j] = s * ap + c * aq;
            }
            __syncthreads();
            // cols: A <- A J
            for (int idx = tid; idx < 64 * DIMN; idx += 128) {
                int k = idx >> 7, j = idx & 127;
                int p = pp[k], q = qq[k]; float c = cs[k], s = sn[k];
                float ap = A[j * STR + p], aq = A[j * STR + q];
                A[j * STR + p] = c * ap - s * aq;
                A[j * STR + q] = s * ap + c * aq;
            }
            __syncthreads();
            // Q <- Q J
            for (int idx = tid; idx < 64 * DIMN; idx += 128) {
                int k = idx >> 7, j = idx & 127;
                int p = pp[k], q = qq[k]; float c = cs[k], s = sn[k];
                float ap = Q[j * STR + p], aq = Q[j * STR + q];
                Q[j * STR + p] = c * ap - s * aq;
                Q[j * STR + q] = s * ap + c * aq;
            }
            __syncthreads();
        }
    }

    if (tid < DIMN) lamg[tid] = A[tid * STR + tid];

    // ---- V = Q^T L0^-1 : row c solves L0^T y = Q[:,c]  (backward substitution) ----
    {
        const int c = tid;
        float y[DIMN];
        for (int i = DIMN - 1; i >= 0; --i) {
            float s = Q[i * STR + c];
            for (int j = i + 1; j < DIMN; ++j) s -= L0[j * STR + i] * y[j];
            y[i] = s / L0[i * STR + i];
        }
        float e = 0.f;
        for (int i = 0; i < DIMN; ++i) {
            Vg[c * DIMN + i] = y[i];
            e += y[i] * (mean1[i] - mean0[i]);
        }
        eg[c] = e;
    }
    if (tid == 0) {
        float s = 0.f;
        for (int i = 0; i < DIMN; ++i) s += logf(L0[i * STR + i]);
        *slogg = s;
    }
}

// ---------------------------------------------------------------------------
// Fused batch kernel: 128 threads = 4 waves; each wave owns 16 samples.
// GEMM1: Y = D * V^T (fp32 WMMA 16x16x4), elementwise w = y/d + reductions,
// GEMM2: G = W * V, writes logp / dxb / dbeta / dt.
// ---------------------------------------------------------------------------
__global__ void lgc_fused(const float* __restrict__ xb,
                          const float* __restrict__ t,
                          const float* __restrict__ mean0,
                          const float* __restrict__ mean1,
                          const float* __restrict__ Vg,
                          const float* __restrict__ lamg,
                          const float* __restrict__ eg,
                          const float* __restrict__ slogg,
                          float* __restrict__ out)
{
    extern __shared__ float sm[];
    float* Vs     = sm;                          // 128*STR
    float* lamS   = Vs + DIMN * STR;             // 128
    float* eS     = lamS + DIMN;                 // 128
    float* Dt_all = eS + DIMN;                   // 4 * 16*STR
    float* Wt_all = Dt_all + 4 * 16 * STR;       // 4 * 16*STR
    float* scal   = Wt_all + 4 * 16 * STR;       // 4 * 48

    const int tid  = threadIdx.x;                // 128
    const int wv   = tid >> 5;
    const int lane = tid & 31;
    const int half = lane >> 4;
    const int l15  = lane & 15;
    const int s0   = blockIdx.x * 64 + wv * 16;  // first sample of this wave's tile

    float* Dt = Dt_all + wv * 16 * STR;
    float* Wt = Wt_all + wv * 16 * STR;
    float* sc = scal + wv * 48;                  // [0:16)=q [16:32)=e [32:48)=l

    for (int idx = tid; idx < DIMN * DIMN; idx += 128)
        Vs[(idx >> 7) * STR + (idx & 127)] = Vg[idx];
    if (tid < DIMN) { lamS[tid] = lamg[tid]; eS[tid] = eg[tid]; }
    __syncthreads();
    const float slog0 = *slogg;

    // per-lane row metadata (M rows this lane's accumulators cover)
    float tm[8], bm[8];
#pragma unroll
    for (int v = 0; v < 8; ++v) {
        int samp = s0 + v + 8 * half;
        tm[v] = t[samp];
        bm[v] = xb[samp * 129 + 128];
    }

    // build D tile: D[s][k] = x - mu(t)
    for (int idx = lane; idx < 16 * DIMN; idx += 32) {
        int s = idx >> 7, k = idx & 127;
        int samp = s0 + s;
        float tt = t[samp];
        Dt[s * STR + k] = xb[samp * 129 + k] - ((1.f - tt) * mean0[k] + tt * mean1[k]);
    }
    __syncthreads();

    float qacc[8], eacc[8], lacc[8];
#pragma unroll
    for (int v = 0; v < 8; ++v) { qacc[v] = 0.f; eacc[v] = 0.f; lacc[v] = 0.f; }

    // ---- GEMM1: Y[m][r] = sum_k D[m][k] V[r][k] ----
    for (int n = 0; n < 8; ++n) {
        v8f acc = {0.f, 0.f, 0.f, 0.f, 0.f, 0.f, 0.f, 0.f};
        for (int kk = 0; kk < 32; ++kk) {
            int k0 = kk * 4 + 2 * half;
            v2f a, b;
            a.x = Dt[l15 * STR + k0];
            a.y = Dt[l15 * STR + k0 + 1];
            b.x = Vs[(n * 16 + l15) * STR + k0];        // B[k][col] = V[col][k]
            b.y = Vs[(n * 16 + l15) * STR + k0 + 1];
            acc = __builtin_amdgcn_wmma_f32_16x16x4_f32(
                false, a, false, b, (short)0, acc, false, false);
        }
#pragma unroll
        for (int v = 0; v < 8; ++v) {
            int m = v + 8 * half;
            int feat = n * 16 + l15;
            float la  = lamS[feat];
            float den = (1.f - tm[v]) + tm[v] * la;
            float y = acc[v];
            float w = y / den;
            Wt[m * STR + feat] = w;
            qacc[v] += y * w;
            eacc[v] += eS[feat] * w;
            lacc[v] += (la - 1.f) * w * w;
        }
    }
    // reduce over the 16 lanes of each half (N dimension)
#pragma unroll
    for (int v = 0; v < 8; ++v) {
        for (int mask = 1; mask <= 8; mask <<= 1) {
            qacc[v] += __shfl_xor(qacc[v], mask, 32);
            eacc[v] += __shfl_xor(eacc[v], mask, 32);
            lacc[v] += __shfl_xor(lacc[v], mask, 32);
        }
    }
    if (l15 == 0) {
#pragma unroll
        for (int v = 0; v < 8; ++v) {
            int m = v + 8 * half;
            sc[m] = qacc[v]; sc[16 + m] = eacc[v]; sc[32 + m] = lacc[v];
        }
    }
    __syncthreads();

    // ---- per-sample scalars: logp, dbeta, dt ----
    if (lane < 16) {
        int m = lane, samp = s0 + m;
        float tt   = t[samp];
        float beta = xb[samp * 129 + 128];
        float sld = 0.f, str_ = 0.f;
        for (int f = 0; f < DIMN; ++f) {
            float la  = lamS[f];
            float den = (1.f - tt) + tt * la;
            sld  += logf(den);
            str_ += (la - 1.f) / den;
        }
        float q = sc[m], ee = sc[16 + m], ll = sc[32 + m];
        out[samp * 131 + 0]   = -0.5f * beta * q - 0.5f * sld - slog0
                                + 64.f * logf(beta) - 64.f * LOG2PI_F;
        out[samp * 131 + 129] = -0.5f * q + 64.f / beta;
        out[samp * 131 + 130] = beta * ee + 0.5f * beta * ll - 0.5f * str_;
    }
    __syncthreads();

    // ---- GEMM2: G[m][k] = sum_r W[m][r] V[r][k];  dx = -beta * G ----
    for (int n = 0; n < 8; ++n) {
        v8f acc = {0.f, 0.f, 0.f, 0.f, 0.f, 0.f, 0.f, 0.f};
        for (int kk = 0; kk < 32; ++kk) {
            int k0 = kk * 4 + 2 * half;
            v2f a, b;
            a.x = Wt[l15 * STR + k0];
            a.y = Wt[l15 * STR + k0 + 1];
            b.x = Vs[k0 * STR + n * 16 + l15];          // B[r][col] = V[r][col]
            b.y = Vs[(k0 + 1) * STR + n * 16 + l15];
            acc = __builtin_amdgcn_wmma_f32_16x16x4_f32(
                false, a, false, b, (short)0, acc, false, false);
        }
#pragma unroll
        for (int v = 0; v < 8; ++v) {
            int samp = s0 + v + 8 * half;
            int feat = n * 16 + l15;
            out[samp * 131 + 1 + feat] = -bm[v] * acc[v];
        }
    }
}

extern "C" void kernel_launch(void* const* d_in, const int* in_sizes, int n_in,
                              void* d_out, int out_size, void* d_ws, size_t ws_size,
                              hipStream_t stream) {
    (void)in_sizes; (void)n_in; (void)out_size; (void)ws_size;
    const float* xb = (const float*)d_in[0];   // (8192,129)
    const float* t  = (const float*)d_in[1];   // (8192,1)
    const float* m0 = (const float*)d_in[2];   // (128,)
    const float* c0 = (const float*)d_in[3];   // (128,128)
    const float* m1 = (const float*)d_in[4];   // (128,)
    const float* c1 = (const float*)d_in[5];   // (128,128)
    float* out = (float*)d_out;                // (8192,131)

    float* ws    = (float*)d_ws;
    float* Vg    = ws;               // 16384
    float* lamg  = ws + 16384;       // 128
    float* eg    = ws + 16512;       // 128
    float* slogg = ws + 16640;       // 1

    size_t sh1 = (size_t)(3 * DIMN * STR) * sizeof(float);                       // ~198 KB
    hipLaunchKernelGGL(lgc_precompute, dim3(1), dim3(128), sh1, stream,
                       c0, c1, m0, m1, Vg, lamg, eg, slogg);

    size_t sh2 = (size_t)(DIMN * STR + 2 * DIMN + 2 * 4 * 16 * STR + 4 * 48) * sizeof(float); // ~134 KB
    hipLaunchKernelGGL(lgc_fused, dim3(8192 / 64), dim3(128), sh2, stream,
                       xb, t, m0, m1, Vg, lamg, eg, slogg, out);
}